// CachedSelfAttention_17686675325444
// MI455X (gfx1250) — compile-verified
//
#include <hip/hip_runtime.h>
#include <cstddef>

// CachedSelfAttention for MI455X (gfx1250, wave32).
// Pipeline:  A: qkv = x@w_qkv+b  (WMMA f32 16x16x4) -> q_ws, k_new/v_new into d_out
//            B: scores = q@K^T * scale, fused cache_k -> k_out copy (NT streaming + LDS stage)
//            C: softmax rows
//            D: ctx = att@V, fused cache_v -> v_out copy, S split in 4 segments (partials)
//            E: out = (sum of ctx partials) @ w_out + b  (WMMA)
// Traffic ~2.15 GB => ~92us roofline @ 23.3 TB/s; all GEMMs on V_WMMA_F32_16X16X4_F32.

#define C_DIM  1024
#define NQKV   3072
#define BT     128
#define B_SZ   16
#define T_SZ   8
#define S_PAST 8192
#define S_TOT  8200
#define SCALE  0.03125f   // 1/sqrt(1024)
#define SBLK   112

typedef __attribute__((ext_vector_type(2))) float v2f;
typedef __attribute__((ext_vector_type(4))) float v4f;
typedef __attribute__((ext_vector_type(8))) float v8f;

static __device__ __forceinline__ v8f wmma_f32(v2f a, v2f b, v8f c) {
  // D(16x16) = A(16x4) * B(4x16) + C  -- f32 WMMA, 8-arg form
  return __builtin_amdgcn_wmma_f32_16x16x4_f32(false, a, false, b, (short)0, c,
                                               false, false);
}

static __device__ __forceinline__ int imin(int a, int b) { return a < b ? a : b; }

// ---------------------------------------------------------------------------
// Kernel A: qkv projection. M=128 (b*t), N=3072, K=1024.
// One 16x16 tile per wave; 8 waves / workgroup; 192 workgroups.
// ---------------------------------------------------------------------------
__global__ __launch_bounds__(256) void qkv_kernel(
    const float* __restrict__ x, const float* __restrict__ w,
    const float* __restrict__ bias, float* __restrict__ q_ws,
    float* __restrict__ k_out, float* __restrict__ v_out) {
  const int wave = threadIdx.x >> 5;
  const int lane = threadIdx.x & 31;
  const int lm = lane & 15, lh = lane >> 4;
  const int tile = blockIdx.x * 8 + wave;         // 0..1535
  const int m0 = (tile & 7) * 16;                 // M tile
  const int n0 = (tile >> 3) * 16;                // N tile
  const float* arow = x + (size_t)(m0 + lm) * C_DIM + 2 * lh;
  const float* bcol = w + (size_t)(2 * lh) * NQKV + n0 + lm;
  v8f acc = {};
  for (int c0 = 0; c0 < C_DIM; c0 += 4) {
    v2f a = *(const v2f*)(arow + c0);             // A[lm][c0+2lh .. +1]
    v2f bb;
    bb.x = bcol[(size_t)c0 * NQKV];               // B[c0+2lh][n]
    bb.y = bcol[(size_t)c0 * NQKV + NQKV];        // B[c0+2lh+1][n]
    acc = wmma_f32(a, bb, acc);
  }
  const int n = n0 + lm;
  const float bv = bias[n];
#pragma unroll
  for (int j = 0; j < 8; ++j) {
    int m = m0 + j + 8 * lh;                      // D row
    int b = m >> 3, t = m & 7;
    float val = acc[j] + bv;
    if (n < C_DIM)
      q_ws[(size_t)m * C_DIM + n] = val;
    else if (n < 2 * C_DIM)
      k_out[((size_t)b * S_TOT + S_PAST + t) * C_DIM + (n - C_DIM)] = val;
    else
      v_out[((size_t)b * S_TOT + S_PAST + t) * C_DIM + (n - 2 * C_DIM)] = val;
  }
}

// ---------------------------------------------------------------------------
// Kernel B: scores[b,t,s] = (q . k_s) * scale, fused with cache_k -> k_out copy.
// Workgroup = (s-macrotile of 128, batch); wave handles 16 s-rows.
// K rows staged through LDS (read cache_k exactly once, NT-streamed).
// ---------------------------------------------------------------------------
__global__ __launch_bounds__(256) void attn_scores_kernel(
    const float* __restrict__ cache_k, const float* __restrict__ q_ws,
    float* __restrict__ k_out, float* __restrict__ scores) {
  __shared__ float ks[8][16 * 68];                // 34 KB, per-wave tiles
  const int b = blockIdx.y;
  const int sm = blockIdx.x;                      // 0..64
  const int wave = threadIdx.x >> 5;
  const int lane = threadIdx.x & 31;
  const int lm = lane & 15, lh = lane >> 4;
  const int s0 = sm * 128 + wave * 16;
  if (s0 >= S_TOT) return;                        // uniform per wave; no barriers used
  const float* kc = cache_k + (size_t)b * S_PAST * C_DIM;
  float* ko = k_out + (size_t)b * S_TOT * C_DIM;
  const float* qb = q_ws + (size_t)b * T_SZ * C_DIM;
  float* myk = &ks[wave][0];
  const v2f zq = {};
  v8f acc = {};
  for (int cc = 0; cc < C_DIM; cc += 64) {
    // Stage 16 rows x 64 cols; simultaneously stream the copy to k_out.
#pragma unroll
    for (int i = 0; i < 8; ++i) {
      int r = i * 2 + lh;                         // 0..15
      int s = s0 + r;
      int ss = s < S_TOT ? s : (S_TOT - 1);
      int col = cc + lm * 4;
      const float* src = (ss < S_PAST) ? (kc + (size_t)ss * C_DIM + col)
                                       : (ko + (size_t)ss * C_DIM + col);
      v4f d = __builtin_nontemporal_load((const v4f*)src);
      if (s < S_PAST)
        __builtin_nontemporal_store(d, (v4f*)(ko + (size_t)s * C_DIM + col));
      *(v4f*)(myk + r * 68 + lm * 4) = d;         // 16B aligned (68*4=272)
    }
    // WMMA: A = K tile (M=s), B = Q^T (K=c, N=t). Q rows read from L2.
#pragma unroll 4
    for (int c4 = 0; c4 < 64; c4 += 4) {
      int c0 = cc + c4;
      v2f a = *(const v2f*)(myk + lm * 68 + c4 + 2 * lh);
      v2f q = (lm < T_SZ) ? *(const v2f*)(qb + (size_t)lm * C_DIM + c0 + 2 * lh)
                          : zq;
      acc = wmma_f32(a, q, acc);
    }
  }
  if (lm < T_SZ) {
    float* srow = scores + ((size_t)b * T_SZ + lm) * S_TOT;
#pragma unroll
    for (int j = 0; j < 8; ++j) {
      int s = s0 + j + 8 * lh;
      if (s < S_TOT) srow[s] = acc[j] * SCALE;
    }
  }
}

// ---------------------------------------------------------------------------
// Kernel C: row softmax over S (in place). One workgroup per (b,t) row.
// ---------------------------------------------------------------------------
__global__ __launch_bounds__(256) void softmax_kernel(float* __restrict__ scores) {
  __shared__ float red[256];
  float* p = scores + (size_t)blockIdx.x * S_TOT;
  const int tid = threadIdx.x;
  float mx = -3.402823466e+38f;
  for (int i = tid; i < S_TOT; i += 256) mx = fmaxf(mx, p[i]);
  red[tid] = mx;
  __syncthreads();
  for (int off = 128; off > 0; off >>= 1) {
    if (tid < off) red[tid] = fmaxf(red[tid], red[tid + off]);
    __syncthreads();
  }
  mx = red[0];
  __syncthreads();
  float sum = 0.0f;
  for (int i = tid; i < S_TOT; i += 256) {
    float e = __expf(p[i] - mx);
    p[i] = e;
    sum += e;
  }
  red[tid] = sum;
  __syncthreads();
  for (int off = 128; off > 0; off >>= 1) {
    if (tid < off) red[tid] += red[tid + off];
    __syncthreads();
  }
  float inv = 1.0f / red[0];
  for (int i = tid; i < S_TOT; i += 256) p[i] *= inv;
}

// ---------------------------------------------------------------------------
// Kernel D: ctx_part[seg] = att[:, seg-range] @ V[seg-range, :] fused with
// cache_v -> v_out copy. Grid (cm=8 col-macrotiles, b=16, seg=4).
// V block staged in LDS; each wave owns a 16-wide column slice.
// ---------------------------------------------------------------------------
__global__ __launch_bounds__(256) void attn_ctx_kernel(
    const float* __restrict__ cache_v, const float* __restrict__ att,
    float* __restrict__ v_out, float* __restrict__ ctx_part) {
  __shared__ float vls[SBLK * 132];               // ~58 KB
  const int cm = blockIdx.x;                      // 0..7  (128 cols each)
  const int b = blockIdx.y;
  const int seg = blockIdx.z;                     // 0..3
  const int seg_lo = seg * 2048;
  const int seg_hi = (seg == 3) ? S_TOT : seg_lo + 2048;
  const int wave = threadIdx.x >> 5;
  const int lane = threadIdx.x & 31;
  const int lm = lane & 15, lh = lane >> 4;
  const int tid = threadIdx.x;
  const float* vc = cache_v + (size_t)b * S_PAST * C_DIM + cm * 128;
  float* vo = v_out + (size_t)b * S_TOT * C_DIM + cm * 128;
  const float* arow =
      (lm < T_SZ) ? (att + ((size_t)b * T_SZ + lm) * S_TOT + 2 * lh) : att;
  const v2f za = {};
  v8f acc = {};
  for (int sb = seg_lo; sb < seg_hi; sb += SBLK) {
    const int rows = imin(SBLK, seg_hi - sb);     // multiple of 4
    __syncthreads();
    // Stage rows x 128 cols; stream the copy to v_out with NT hints.
    for (int r0 = tid >> 5; r0 < rows; r0 += 8) {
      int s = sb + r0;
      int col = (tid & 31) * 4;
      v4f d;
      if (s < S_PAST) {
        d = __builtin_nontemporal_load((const v4f*)(vc + (size_t)s * C_DIM + col));
        __builtin_nontemporal_store(d, (v4f*)(vo + (size_t)s * C_DIM + col));
        if (s + SBLK < S_PAST)
          __builtin_prefetch(vc + (size_t)(s + SBLK) * C_DIM + col, 0, 1);
      } else {
        d = *(const v4f*)(vo + (size_t)s * C_DIM + col);  // v_new rows from kernel A
      }
      *(v4f*)(&vls[r0 * 132 + col]) = d;          // 16B aligned (132*4=528)
    }
    __syncthreads();
    // WMMA: A = att tile (M=t, K=s), B = V block (K=s, N=c).
    for (int sr = 0; sr < rows; sr += 4) {
      v2f a = (lm < T_SZ) ? *(const v2f*)(arow + sb + sr) : za;
      v2f vv;
      vv.x = vls[(sr + 2 * lh) * 132 + wave * 16 + lm];
      vv.y = vls[(sr + 2 * lh + 1) * 132 + wave * 16 + lm];
      acc = wmma_f32(a, vv, acc);
    }
  }
  if (lh == 0) {                                  // rows t=0..7 live in lanes 0-15
#pragma unroll
    for (int j = 0; j < 8; ++j)
      ctx_part[((size_t)seg * BT + b * T_SZ + j) * C_DIM + cm * 128 + wave * 16 + lm] =
          acc[j];
  }
}

// ---------------------------------------------------------------------------
// Kernel E: out = (sum_seg ctx_part) @ w_out + b_out. M=128, N=1024, K=1024.
// ---------------------------------------------------------------------------
__global__ __launch_bounds__(256) void outproj_kernel(
    const float* __restrict__ ctx_part, const float* __restrict__ w,
    const float* __restrict__ bias, float* __restrict__ out) {
  const int wave = threadIdx.x >> 5;
  const int lane = threadIdx.x & 31;
  const int lm = lane & 15, lh = lane >> 4;
  const int tile = blockIdx.x * 8 + wave;         // 0..511
  const int m0 = (tile & 7) * 16;
  const int n0 = (tile >> 3) * 16;
  const float* arow = ctx_part + (size_t)(m0 + lm) * C_DIM + 2 * lh;
  const float* bcol = w + (size_t)(2 * lh) * C_DIM + n0 + lm;
  v8f acc = {};
  for (int c0 = 0; c0 < C_DIM; c0 += 4) {
    v2f a = *(const v2f*)(arow + c0);
    a += *(const v2f*)(arow + c0 + 1 * BT * C_DIM);
    a += *(const v2f*)(arow + c0 + 2 * BT * C_DIM);
    a += *(const v2f*)(arow + c0 + 3 * BT * C_DIM);
    v2f bb;
    bb.x = bcol[(size_t)c0 * C_DIM];
    bb.y = bcol[(size_t)c0 * C_DIM + C_DIM];
    acc = wmma_f32(a, bb, acc);
  }
  const int n = n0 + lm;
  const float bv = bias[n];
#pragma unroll
  for (int j = 0; j < 8; ++j) {
    int m = m0 + j + 8 * lh;
    out[(size_t)m * C_DIM + n] = acc[j] + bv;
  }
}

// ---------------------------------------------------------------------------
extern "C" void kernel_launch(void* const* d_in, const int* in_sizes, int n_in,
                              void* d_out, int out_size, void* d_ws, size_t ws_size,
                              hipStream_t stream) {
  const float* x       = (const float*)d_in[0];
  const float* cache_k = (const float*)d_in[1];
  const float* cache_v = (const float*)d_in[2];
  const float* w_qkv   = (const float*)d_in[3];
  const float* b_qkv   = (const float*)d_in[4];
  const float* w_out   = (const float*)d_in[5];
  const float* b_out   = (const float*)d_in[6];

  float* out   = (float*)d_out;                              // [16,8,1024]
  float* k_out = out + (size_t)BT * C_DIM;                   // [16,8200,1024]
  float* v_out = k_out + (size_t)B_SZ * S_TOT * C_DIM;       // [16,8200,1024]

  float* q_ws     = (float*)d_ws;                            // 128*1024
  float* ctx_part = q_ws + (size_t)BT * C_DIM;               // 4*128*1024
  float* scores   = ctx_part + (size_t)4 * BT * C_DIM;       // 16*8*8200

  qkv_kernel<<<192, 256, 0, stream>>>(x, w_qkv, b_qkv, q_ws, k_out, v_out);
  attn_scores_kernel<<<dim3(65, 16), 256, 0, stream>>>(cache_k, q_ws, k_out, scores);
  softmax_kernel<<<BT, 256, 0, stream>>>(scores);
  attn_ctx_kernel<<<dim3(8, 16, 4), 256, 0, stream>>>(cache_v, scores, v_out, ctx_part);
  outproj_kernel<<<64, 256, 0, stream>>>(ctx_part, w_out, b_out, out);
}